// AttentionLayer_85547158602286
// MI455X (gfx1250) — compile-verified
//
#include <hip/hip_runtime.h>

#define B_  8
#define TE_ 512
#define TD_ 512
#define D_  128
#define TG_ 4   // decoder timesteps per energies block (Ws reuse factor)

typedef float v2f __attribute__((ext_vector_type(2)));
typedef float v8f __attribute__((ext_vector_type(8)));

// ---------------------------------------------------------------------------
// Full-precision f32 GEMM tile kernel: one wave32 per 16x16 output tile,
// accumulating with V_WMMA_F32_16X16X4_F32.
//   C[m,n] = sum_k A[m,k] * Bm[k,n]
// A row stride = K, B row stride = N, C row stride = N.
// blockIdx.x -> M tile, blockIdx.y -> N tile, blockIdx.z -> batch (strides).
// Per-lane fragment layout (lane L, reg v), per CDNA5 ISA 7.12.2:
//   A: [M = L%16, K = v + 2*(L/16)]   (2 VGPRs)
//   B: [K = v + 2*(L/16), N = L%16]   (2 VGPRs)
//   C: [M = v + 8*(L/16), N = L%16]   (8 VGPRs)
// ---------------------------------------------------------------------------
__global__ void gemm16_wmma_f32(const float* __restrict__ A,
                                const float* __restrict__ Bm,
                                float* __restrict__ C,
                                int K, int N,
                                long long sA, long long sB, long long sC) {
    const int lane = threadIdx.x;       // 0..31, block is exactly one wave
    const int lid  = lane & 15;
    const int half = lane >> 4;         // 0 or 1
    const int m0   = blockIdx.x * 16;
    const int n0   = blockIdx.y * 16;

    A  += (long long)blockIdx.z * sA;
    Bm += (long long)blockIdx.z * sB;
    C  += (long long)blockIdx.z * sC;

    const float* arow = A + (long long)(m0 + lid) * K + 2 * half; // 8B aligned
    const float* bcol = Bm + (long long)(2 * half) * N + n0 + lid;

    v8f acc = {};
    #pragma unroll 8
    for (int k0 = 0; k0 < K; k0 += 4) {
        v2f a = *(const v2f*)(arow + k0);          // A[m, k0+2h], A[m, k0+2h+1]
        v2f b;
        b.x = bcol[(long long)k0 * N];             // B[k0+2h,   n]
        b.y = bcol[(long long)(k0 + 1) * N];       // B[k0+2h+1, n]
        acc = __builtin_amdgcn_wmma_f32_16x16x4_f32(
                  false, a, false, b, (short)0, acc, false, false);
    }

    float* crow = C + (long long)(m0 + 8 * half) * N + n0 + lid;
    #pragma unroll
    for (int v = 0; v < 8; ++v)
        crow[(long long)v * N] = acc[v];
}

// ---------------------------------------------------------------------------
// tanh: single V_TANH_F32 (TRANS32) when available on gfx1250, else
// exp2-based: t = exp2(-2*log2e*|x|); r = (1-t)/(1+t); copysign.
// ---------------------------------------------------------------------------
__device__ __forceinline__ float tanh_fast(float x) {
#if __has_builtin(__builtin_amdgcn_tanhf)
    return __builtin_amdgcn_tanhf(x);
#else
    const float k = -2.885390082f;                  // -2 * log2(e)
    float t = __builtin_amdgcn_exp2f(k * fabsf(x)); // |x| is a free src modifier
    float r = (1.0f - t) * __builtin_amdgcn_rcpf(1.0f + t);
    return copysignf(r, x);
#endif
}

// ---------------------------------------------------------------------------
// Energies + softmax, fused. One 256-thread block per (b, group of TG_ t's):
//   e[b,t,s] = softmax_s( sum_d tanh(Ws[b,s,d] + Uh[b,t,d]) * V[d] )
// Each Ws float4 loaded from L2 is reused for TG_ decoder rows (register
// reuse), cutting aggregate L2 read traffic from ~1 GB to ~256 MB. Uh rows
// and V live in LDS. Each thread owns 2 encoder positions per t.
// ---------------------------------------------------------------------------
__global__ void energies_softmax_kernel(const float* __restrict__ Ws,
                                        const float* __restrict__ Uh,
                                        const float* __restrict__ Va,
                                        float* __restrict__ e_out) {
    __shared__ float4 s_uh[TG_][D_ / 4];
    __shared__ float4 s_v[D_ / 4];
    __shared__ float  s_red[256];

    const int tid   = threadIdx.x;
    const int b     = blockIdx.x / (TD_ / TG_);
    const int tbase = (blockIdx.x % (TD_ / TG_)) * TG_;

    if (tid < TG_ * (D_ / 4)) {
        const int tt = tid / (D_ / 4);
        const int i  = tid % (D_ / 4);
        s_uh[tt][i] = ((const float4*)(Uh + (size_t)(b * TD_ + tbase + tt) * D_))[i];
    }
    if (tid < D_ / 4) s_v[tid] = ((const float4*)Va)[tid];
    __syncthreads();

    float acc[2][TG_];
    #pragma unroll
    for (int r = 0; r < 2; ++r)
        #pragma unroll
        for (int tt = 0; tt < TG_; ++tt) acc[r][tt] = 0.0f;

    #pragma unroll
    for (int r = 0; r < 2; ++r) {
        const int s = tid + r * 256;
        const float4* wrow = (const float4*)(Ws + (size_t)(b * TE_ + s) * D_);
        #pragma unroll 4
        for (int i = 0; i < D_ / 4; ++i) {
            const float4 w = wrow[i];       // one global load, reused TG_ times
            const float4 v = s_v[i];
            #pragma unroll
            for (int tt = 0; tt < TG_; ++tt) {
                const float4 u = s_uh[tt][i];
                float a = acc[r][tt];
                a = fmaf(tanh_fast(w.x + u.x), v.x, a);
                a = fmaf(tanh_fast(w.y + u.y), v.y, a);
                a = fmaf(tanh_fast(w.z + u.z), v.z, a);
                a = fmaf(tanh_fast(w.w + u.w), v.w, a);
                acc[r][tt] = a;
            }
        }
    }

    // per-t softmax over the 512 encoder positions
    #pragma unroll
    for (int tt = 0; tt < TG_; ++tt) {
        s_red[tid] = fmaxf(acc[0][tt], acc[1][tt]);
        __syncthreads();
        for (int s = 128; s > 0; s >>= 1) {
            if (tid < s) s_red[tid] = fmaxf(s_red[tid], s_red[tid + s]);
            __syncthreads();
        }
        const float mx = s_red[0];
        __syncthreads();

        const float ex0 = __expf(acc[0][tt] - mx);
        const float ex1 = __expf(acc[1][tt] - mx);
        s_red[tid] = ex0 + ex1;
        __syncthreads();
        for (int s = 128; s > 0; s >>= 1) {
            if (tid < s) s_red[tid] += s_red[tid + s];
            __syncthreads();
        }
        const float inv = 1.0f / s_red[0];
        __syncthreads();   // s_red reused next tt iteration

        float* orow = e_out + (size_t)(b * TD_ + tbase + tt) * TE_;
        orow[tid]       = ex0 * inv;
        orow[tid + 256] = ex1 * inv;
    }
}

// ---------------------------------------------------------------------------
// Launch: Ws/Uh GEMMs -> energies+softmax -> context GEMM (batched).
// d_out = [c (B*TD*D) | e (B*TD*TE)], d_ws = [Ws (2MB) | Uh (2MB)].
// ---------------------------------------------------------------------------
extern "C" void kernel_launch(void* const* d_in, const int* in_sizes, int n_in,
                              void* d_out, int out_size, void* d_ws, size_t ws_size,
                              hipStream_t stream) {
    const float* enc = (const float*)d_in[0];   // [B, TE, D]
    const float* dec = (const float*)d_in[1];   // [B, TD, D]
    const float* Wa  = (const float*)d_in[2];   // [D, D]
    const float* Ua  = (const float*)d_in[3];   // [D, D]
    const float* Va  = (const float*)d_in[4];   // [D, 1]

    float* c_out = (float*)d_out;                          // [B, TD, D]
    float* e_out = (float*)d_out + (size_t)B_ * TD_ * D_;  // [B, TD, TE]

    float* Ws = (float*)d_ws;                              // [B*TE, D]
    float* Uh = Ws + (size_t)B_ * TE_ * D_;                // [B*TD, D]

    // Ws = enc @ Wa  : [4096 x 128] * [128 x 128]
    gemm16_wmma_f32<<<dim3((B_ * TE_) / 16, D_ / 16, 1), 32, 0, stream>>>(
        enc, Wa, Ws, D_, D_, 0, 0, 0);

    // Uh = dec @ Ua  : [4096 x 128] * [128 x 128]
    gemm16_wmma_f32<<<dim3((B_ * TD_) / 16, D_ / 16, 1), 32, 0, stream>>>(
        dec, Ua, Uh, D_, D_, 0, 0, 0);

    // e = softmax(V^T tanh(Ws + Uh)) : one block per (b, 4 t's)
    energies_softmax_kernel<<<B_ * (TD_ / TG_), 256, 0, stream>>>(Ws, Uh, Va, e_out);

    // c[b] = e[b] @ enc[b] : batched [512 x 512] * [512 x 128]
    gemm16_wmma_f32<<<dim3(TD_ / 16, D_ / 16, B_), 32, 0, stream>>>(
        e_out, enc, c_out, TE_, D_,
        (long long)TD_ * TE_, (long long)TE_ * D_, (long long)TD_ * D_);
}